// ContinuousConv_46291157517027
// MI455X (gfx1250) — compile-verified
//
#include <hip/hip_runtime.h>

typedef __attribute__((ext_vector_type(2))) float v2f;
typedef __attribute__((ext_vector_type(8))) float v8f;

#define NPTS   200000
#define MOUT   100000
#define CIN    32
#define COUT   64
#define KNBR   32
#define VOX    27
#define KDIM   (VOX * CIN)      // 864  (GEMM reduction depth)
#define AS     (KDIM + 2)       // LDS row stride (dwords): 866 -> conflict-free, 8B aligned cols
#define MTILE  16
#define WAVES  4
#define BLOCK  (WAVES * 32)

__global__ __launch_bounds__(BLOCK)
void cconv_fused_kernel(const float* __restrict__ feats,      // [N, CIN]
                        const float* __restrict__ inPos,      // [N, 3]
                        const float* __restrict__ outPos,     // [M, 3]
                        const float* __restrict__ extents,    // [1]
                        const float* __restrict__ importance, // [N]
                        const int*   __restrict__ nbrIdx,     // [M, K]
                        const float* __restrict__ Wg,         // [3,3,3,CIN,COUT] == [864, 64]
                        const float* __restrict__ bias,       // [COUT]
                        float*       __restrict__ out)        // [M, COUT]
{
    __shared__ float sA[MTILE][AS];   // A matrix tile: 16 x 864 (padded)
    __shared__ float sInvN[MTILE];    // per-row 1/norm

    const int lane = threadIdx.x & 31;   // lane == input channel in phase 1
    const int wave = threadIdx.x >> 5;
    const int m0   = blockIdx.x * MTILE;
    const float invR = 2.0f / extents[0];        // 1 / (0.5 * extents[0])

    // ---------------- Phase 1: build acc[16, 27, 32] (lane = cin) ----------------
    for (int r = 0; r < MTILE / WAVES; ++r) {
        const int row = wave * (MTILE / WAVES) + r;
        const int m   = m0 + row;
        const float ox = outPos[m * 3 + 0];
        const float oy = outPos[m * 3 + 1];
        const float oz = outPos[m * 3 + 2];

        float acc[VOX];
        #pragma unroll
        for (int v = 0; v < VOX; ++v) acc[v] = 0.0f;
        float norm = 0.0f;

        for (int k = 0; k < KNBR; ++k) {
            const int idx = nbrIdx[m * KNBR + k];              // wave-uniform
            const float rx = (inPos[idx * 3 + 0] - ox) * invR;
            const float ry = (inPos[idx * 3 + 1] - oy) * invR;
            const float rz = (inPos[idx * 3 + 2] - oz) * invR;
            const float dist = sqrtf(rx * rx + ry * ry + rz * rz);
            const float linf = fmaxf(fabsf(rx), fmaxf(fabsf(ry), fabsf(rz)));
            const float scale = (linf > 1e-8f) ? dist / fmaxf(linf, 1e-8f) : 0.0f;
            const float imp = (dist <= 1.0f) ? importance[idx] : 0.0f;

            if (imp != 0.0f) {                                 // wave-uniform branch
                norm += imp;
                // ball->cube, then map [-1,1] -> [0,2] voxel-center coords
                const float cx = fminf(fmaxf(rx * scale + 1.0f, 0.0f), 2.0f);
                const float cy = fminf(fmaxf(ry * scale + 1.0f, 0.0f), 2.0f);
                const float cz = fminf(fmaxf(rz * scale + 1.0f, 0.0f), 2.0f);
                float wx[3], wy[3], wz[3];
                #pragma unroll
                for (int a = 0; a < 3; ++a) {
                    wx[a] = fmaxf(0.0f, 1.0f - fabsf(cx - (float)a));
                    wy[a] = fmaxf(0.0f, 1.0f - fabsf(cy - (float)a));
                    wz[a] = fmaxf(0.0f, 1.0f - fabsf(cz - (float)a));
                }
                const float f = feats[idx * CIN + lane] * imp; // coalesced 128B gather (L2)
                const float fz0 = wz[0] * f, fz1 = wz[1] * f, fz2 = wz[2] * f;
                #pragma unroll
                for (int a = 0; a < 3; ++a) {
                    #pragma unroll
                    for (int b = 0; b < 3; ++b) {
                        const float wab = wx[a] * wy[b];
                        acc[(a * 3 + b) * 3 + 0] = fmaf(wab, fz0, acc[(a * 3 + b) * 3 + 0]);
                        acc[(a * 3 + b) * 3 + 1] = fmaf(wab, fz1, acc[(a * 3 + b) * 3 + 1]);
                        acc[(a * 3 + b) * 3 + 2] = fmaf(wab, fz2, acc[(a * 3 + b) * 3 + 2]);
                    }
                }
            }
        }

        #pragma unroll
        for (int v = 0; v < VOX; ++v)
            sA[row][v * CIN + lane] = acc[v];                  // column = voxel*CIN + cin
        if (lane == 0)
            sInvN[row] = (norm > 0.0f) ? 1.0f / fmaxf(norm, 1e-12f) : 0.0f;
    }
    __syncthreads();

    // ---------------- Phase 2: C[16x16] = A[16x864] * W[864x16] per wave --------
    // f32 WMMA 16x16x4.  A layout: lanes 0-15 M=0..15 {K+0,K+1}, lanes 16-31 {K+2,K+3}.
    // B layout (mirror):  lanes 0-15 rows {K+0,K+1} col N=lane, lanes 16-31 rows {K+2,K+3}.
    const int n0   = wave * 16;              // COUT tile base
    const int half = lane >> 4;              // 0 or 1
    const int lm   = lane & 15;
    const float* Wt = Wg + n0 + lm;          // column (n0+lm) of the 864x64 filter

    v8f c = {};
    #pragma unroll 4
    for (int kk = 0; kk < KDIM / 4; ++kk) {
        const int kb = kk * 4 + 2 * half;
        v2f a;
        a.x = sA[lm][kb + 0];                // merges to ds_load_b64 (8B aligned)
        a.y = sA[lm][kb + 1];
        v2f b;
        b.x = Wt[(kb + 0) * COUT];           // L2-broadcast filter reads
        b.y = Wt[(kb + 1) * COUT];
        c = __builtin_amdgcn_wmma_f32_16x16x4_f32(
                /*neg_a=*/false, a, /*neg_b=*/false, b,
                /*c_mod=*/(short)0, c, /*reuse_a=*/false, /*reuse_b=*/false);
    }

    // Fused normalize + bias + store.  C layout: VGPR j, lane L -> M=j+8*(L/16), N=L&15.
    const float bv = bias[n0 + lm];
    #pragma unroll
    for (int j = 0; j < 8; ++j) {
        const int mrow = j + 8 * half;
        out[(long)(m0 + mrow) * COUT + n0 + lm] = c[j] * sInvN[mrow] + bv;
    }
}

extern "C" void kernel_launch(void* const* d_in, const int* in_sizes, int n_in,
                              void* d_out, int out_size, void* d_ws, size_t ws_size,
                              hipStream_t stream) {
    const float* feats      = (const float*)d_in[0];
    const float* inPos      = (const float*)d_in[1];
    const float* outPos     = (const float*)d_in[2];
    const float* extents    = (const float*)d_in[3];
    const float* importance = (const float*)d_in[4];
    const int*   nbrIdx     = (const int*)  d_in[5];
    const float* Wg         = (const float*)d_in[6];
    const float* bias       = (const float*)d_in[7];
    float* out = (float*)d_out;

    dim3 grid(MOUT / MTILE);   // 100000 / 16 = 6250, exact
    cconv_fused_kernel<<<grid, BLOCK, 0, stream>>>(
        feats, inPos, outPos, extents, importance, nbrIdx, Wg, bias, out);
}